// DynamicLinear_69715909149264
// MI455X (gfx1250) — compile-verified
//
#include <hip/hip_runtime.h>

// Problem constants (match reference)
#define B_N    512
#define IN_K   1024
#define OUT_N  1024
#define NCAT   64
#define MAX_MT 32          // worst case: all 512 rows in one category -> 32 m-tiles
#define NT     128         // output columns per block (8 waves x 16)
#define KC     256         // K chunk staged in LDS
#define LDSTR  (KC + 4)    // padded row stride (bank-conflict-free: 260 % 64 == 4)
#define LDS_FLOATS (16 * LDSTR)
#define LDS_BYTES  (LDS_FLOATS * 4 + 16 * 4)

typedef __attribute__((ext_vector_type(2))) float v2f;
typedef __attribute__((ext_vector_type(8))) float v8f;

// ---------------------------------------------------------------------------
// Pass 1: bucket rows by category.  offsets[65], rowlist[512] written to ws.
// Recomputed every call (no persistent state).  Intra-bucket order from
// atomics is irrelevant: each output row's value is order-independent.
// ---------------------------------------------------------------------------
__global__ __launch_bounds__(256)
void dl_group(const int* __restrict__ cat,
              int* __restrict__ offsets,
              int* __restrict__ rowlist) {
    __shared__ int s_cnt[NCAT];
    __shared__ int s_off[NCAT];
    const int tid = threadIdx.x;

    if (tid < NCAT) s_cnt[tid] = 0;
    __syncthreads();

    for (int b = tid; b < B_N; b += 256)
        atomicAdd(&s_cnt[cat[b] & (NCAT - 1)], 1);
    __syncthreads();

    if (tid == 0) {
        int run = 0;
        for (int c = 0; c < NCAT; ++c) { s_off[c] = run; run += s_cnt[c]; }
    }
    __syncthreads();

    if (tid < NCAT) { offsets[tid] = s_off[tid]; s_cnt[tid] = 0; }
    if (tid == 0)   offsets[NCAT] = B_N;
    __syncthreads();

    for (int b = tid; b < B_N; b += 256) {
        int c = cat[b] & (NCAT - 1);
        int pos = atomicAdd(&s_cnt[c], 1);
        rowlist[s_off[c] + pos] = b;
    }
}

// ---------------------------------------------------------------------------
// Pass 2: per (category, m-tile, n-tile) block.  16 gathered x-rows staged in
// LDS; 8 waves each own a 16x16 f32 WMMA accumulator over 16 output columns.
// W[c] is streamed from HBM exactly once per active m-tile of its category.
// ---------------------------------------------------------------------------
__global__ __launch_bounds__(256)
void dl_gemm(const float* __restrict__ x,
             const float* __restrict__ w,
             const float* __restrict__ bias,
             const int*   __restrict__ offsets,
             const int*   __restrict__ rowlist,
             float*       __restrict__ out) {
    extern __shared__ float lds[];
    int* rows = (int*)&lds[LDS_FLOATS];

    const int c    = blockIdx.y / MAX_MT;
    const int mt   = blockIdx.y % MAX_MT;
    const int off0 = offsets[c];
    const int off1 = offsets[c + 1];
    if (mt * 16 >= off1 - off0) return;     // block-uniform early exit

    const int tid  = threadIdx.x;
    const int wid  = tid >> 5;              // wave 0..7
    const int lane = tid & 31;
    const int half = lane >> 4;             // 0: lanes 0-15, 1: lanes 16-31
    const int lm   = lane & 15;

    const int nl = blockIdx.x * NT + wid * 16 + lm;   // this lane's output col

    if (tid < 16) {
        int gi = off0 + mt * 16 + tid;
        rows[tid] = (gi < off1) ? rowlist[gi] : -1;   // -1 => zero-padded row
    }
    __syncthreads();

    v8f acc = {};
    const float* wbase = w + (size_t)c * (IN_K * OUT_N) + nl;

    for (int kc = 0; kc < IN_K; kc += KC) {
        // ---- stage X[rows, kc:kc+KC] into LDS (coalesced float4) ----------
        {
            const int col4  = (tid & 63) * 4;   // 0..252
            const int rbase = tid >> 6;         // 0..3
            #pragma unroll
            for (int j = 0; j < 4; ++j) {
                int r = rbase + 4 * j;
                int grow = rows[r];
                float4 v = make_float4(0.f, 0.f, 0.f, 0.f);
                if (grow >= 0)
                    v = *(const float4*)&x[(size_t)grow * IN_K + kc + col4];
                *(float4*)&lds[r * LDSTR + col4] = v;
            }
        }
        __syncthreads();

        // ---- K loop: v_wmma_f32_16x16x4_f32 -------------------------------
        // A layout: lane lm row M=lm; VGPR pair holds K = k0+2*half, +1
        // B layout: lane lm col N=nl;  VGPR pair holds K = k0+2*half, +1
        const float* wrow = wbase + (size_t)(kc + 2 * half) * OUT_N;
        const float* arow = &lds[lm * LDSTR + 2 * half];
        #pragma unroll 4
        for (int k0 = 0; k0 < KC; k0 += 4) {
            v2f a = *(const v2f*)&arow[k0];     // contiguous b64 LDS read
            v2f b;
            b.x = wrow[0];                      // W[k, nl]
            b.y = wrow[OUT_N];                  // W[k+1, nl]
            acc = __builtin_amdgcn_wmma_f32_16x16x4_f32(
                      false, a, false, b, (short)0, acc, false, false);
            wrow += 4 * OUT_N;
        }
        __syncthreads();
    }

    // ---- epilogue: bias + scatter-store (D layout: VGPR v -> M = v+8*half)
    const float bv = bias[c * OUT_N + nl];
    #pragma unroll
    for (int v = 0; v < 8; ++v) {
        int m    = v + 8 * half;
        int grow = rows[m];
        if (grow >= 0)
            out[(size_t)grow * OUT_N + nl] = acc[v] + bv;
    }
}

// ---------------------------------------------------------------------------
extern "C" void kernel_launch(void* const* d_in, const int* in_sizes, int n_in,
                              void* d_out, int out_size, void* d_ws, size_t ws_size,
                              hipStream_t stream) {
    (void)in_sizes; (void)n_in; (void)out_size; (void)ws_size;

    const float* x    = (const float*)d_in[0];
    const float* wgt  = (const float*)d_in[1];
    const float* bias = (const float*)d_in[2];
    const int*   cat  = (const int*)d_in[3];
    float*       out  = (float*)d_out;

    int* offsets = (int*)d_ws;          // 65 ints
    int* rowlist = offsets + 128;       // 512 ints (offset 512 B, aligned)

    dl_group<<<dim3(1), dim3(256), 0, stream>>>(cat, offsets, rowlist);

    dim3 grid(OUT_N / NT, NCAT * MAX_MT);   // (8, 2048); empty tiles early-out
    dl_gemm<<<grid, dim3(256), LDS_BYTES, stream>>>(x, wgt, bias,
                                                    offsets, rowlist, out);
}